// NLinear_687194767845
// MI455X (gfx1250) — compile-verified
//
#include <hip/hip_runtime.h>

typedef __attribute__((ext_vector_type(2))) float v2f;
typedef __attribute__((ext_vector_type(8))) float v8f;

#define BATCH 8192
#define NFEAT 64
#define DIN   128
#define DOUT  128
#define BM    64      // batch rows per workgroup
#define LDS_S 132     // padded LDS row stride in floats (4 dwords mod 64 banks)

__global__ __launch_bounds__(256) void nlinear_wmma_kernel(
    const float* __restrict__ x,     // [BATCH, NFEAT, DIN]
    const float* __restrict__ w,     // [NFEAT, DIN, DOUT]
    const float* __restrict__ bias,  // [NFEAT, DOUT]
    float* __restrict__ out)         // [BATCH, NFEAT, DOUT]
{
    __shared__ float ldsX[BM * LDS_S];    // x tile, row-major  [m][k]
    __shared__ float ldsW[DOUT * LDS_S];  // W tile, TRANSPOSED [n][k]

    const int f      = blockIdx.y;
    const int tile_b = blockIdx.x * BM;
    const int tid    = threadIdx.x;        // 0..255 (8 waves of 32)
    const int lane   = tid & 31;
    const int wave   = tid >> 5;
    const int l15    = lane & 15;
    const int hi     = lane >> 4;          // 0: lanes 0-15, 1: lanes 16-31

    // ---- stage x tile: BM x DIN, coalesced float4 global loads ----
    {
        const float* xf = x + (size_t)tile_b * (NFEAT * DIN) + (size_t)f * DIN;
        #pragma unroll
        for (int i = 0; i < (BM * DIN / 4) / 256; ++i) {   // 8 iters
            int idx4 = tid + i * 256;        // float4 index
            int r    = idx4 >> 5;            // 32 float4 per row
            int c4   = (idx4 & 31) * 4;
            float4 v = *(const float4*)(xf + (size_t)r * (NFEAT * DIN) + c4);
            *(float4*)(&ldsX[r * LDS_S + c4]) = v;
        }
    }
    // ---- stage W transposed: ldsW[n][k] = w[f][k][n] ----
    {
        const float* wf = w + (size_t)f * (DIN * DOUT);
        #pragma unroll
        for (int i = 0; i < (DIN * DOUT / 4) / 256; ++i) { // 16 iters
            int idx4 = tid + i * 256;
            int k    = idx4 >> 5;            // 32 float4 per k-row
            int n4   = (idx4 & 31) * 4;
            float4 v = *(const float4*)(wf + k * DOUT + n4);
            ldsW[(n4 + 0) * LDS_S + k] = v.x;
            ldsW[(n4 + 1) * LDS_S + k] = v.y;
            ldsW[(n4 + 2) * LDS_S + k] = v.z;
            ldsW[(n4 + 3) * LDS_S + k] = v.w;
        }
    }
    __syncthreads();

    // wave -> 16(M) x 64(N) sub-block: 4 accumulator tiles of 16x16
    const int m0 = (wave >> 1) * 16;   // 0,16,32,48
    const int n0 = (wave & 1) * 64;    // 0 or 64

    // init accumulators from bias (C/D layout: lane -> N, vgpr r -> M = r + 8*hi)
    v8f acc[4];
    #pragma unroll
    for (int nt = 0; nt < 4; ++nt) {
        float bv = bias[f * DOUT + n0 + nt * 16 + l15];
        v8f a = {};
        #pragma unroll
        for (int r = 0; r < 8; ++r) a[r] = bv;
        acc[nt] = a;
    }

    // K loop: 16x16x4 f32 WMMA; A lanes 0-15 hold K=k,k+1 / lanes 16-31 K=k+2,k+3
    const int koff = 2 * hi;
    const int arow = (m0 + l15) * LDS_S;
    #pragma unroll
    for (int k = 0; k < DIN; k += 4) {
        v2f av = *(const v2f*)(&ldsX[arow + k + koff]);
        #pragma unroll
        for (int nt = 0; nt < 4; ++nt) {
            v2f bv = *(const v2f*)(&ldsW[(n0 + nt * 16 + l15) * LDS_S + k + koff]);
            acc[nt] = __builtin_amdgcn_wmma_f32_16x16x4_f32(
                false, av, false, bv, (short)0, acc[nt], false, false);
        }
    }

    // ---- epilogue: store D per C/D VGPR layout ----
    float* of = out + (size_t)(tile_b + m0) * (NFEAT * DOUT) + (size_t)f * DOUT + n0;
    #pragma unroll
    for (int nt = 0; nt < 4; ++nt) {
        #pragma unroll
        for (int r = 0; r < 8; ++r) {
            of[(size_t)(r + 8 * hi) * (NFEAT * DOUT) + nt * 16 + l15] = acc[nt][r];
        }
    }
}

extern "C" void kernel_launch(void* const* d_in, const int* in_sizes, int n_in,
                              void* d_out, int out_size, void* d_ws, size_t ws_size,
                              hipStream_t stream) {
    const float* x    = (const float*)d_in[0];   // [8192,64,128]
    const float* w    = (const float*)d_in[1];   // [64,128,128]
    const float* bias = (const float*)d_in[2];   // [64,128]
    float* out        = (float*)d_out;           // [8192,64,128]
    (void)in_sizes; (void)n_in; (void)out_size; (void)d_ws; (void)ws_size;

    dim3 grid(BATCH / BM, NFEAT);
    nlinear_wmma_kernel<<<grid, 256, 0, stream>>>(x, w, bias, out);
}